// FactGCN_21320217657490
// MI455X (gfx1250) — compile-verified
//
#include <hip/hip_runtime.h>

typedef float v2f __attribute__((ext_vector_type(2)));
typedef float v8f __attribute__((ext_vector_type(8)));

#define DFEAT 128

// ---------------------------------------------------------------------------
// Repack weight W[K x 128] into WMMA B-fragment order:
//   Bp[(g*128 + c)*2 + h] = { W[(4g+2h)  ][c],
//                             W[(4g+2h+1)][c] }        (float2)
// For k-group g, lane (h*16+m) of wave w loads one contiguous b64 at
// ((g*128 + w*16 + m)*2 + h). One wave's fragment = 256 contiguous bytes.
// ---------------------------------------------------------------------------
__global__ void __launch_bounds__(256)
fgcn_pack_w(const float* __restrict__ W, float2* __restrict__ Bp, int K) {
    const int total = (K / 4) * DFEAT * 2;
    const int idx = blockIdx.x * 256 + threadIdx.x;
    if (idx >= total) return;
    const int h = idx & 1;
    const int c = (idx >> 1) & (DFEAT - 1);
    const int g = idx >> 8;
    float2 v;
    v.x = W[(size_t)(4 * g + 2 * h) * DFEAT + c];
    v.y = W[(size_t)(4 * g + 2 * h + 1) * DFEAT + c];
    Bp[idx] = v;
}

// ---------------------------------------------------------------------------
// zero int workspace
// ---------------------------------------------------------------------------
__global__ void __launch_bounds__(256)
fgcn_zero_i32(int* __restrict__ p, int n) {
    int i = blockIdx.x * 256 + threadIdx.x;
    if (i < n) p[i] = 0;
}

// ---------------------------------------------------------------------------
// CSR build step 1: histogram of destination nodes
// ---------------------------------------------------------------------------
__global__ void __launch_bounds__(256)
fgcn_histogram(const int* __restrict__ dst, int* __restrict__ counts, int E) {
    int e = blockIdx.x * 256 + threadIdx.x;
    if (e < E) atomicAdd(&counts[dst[e]], 1);
}

// ---------------------------------------------------------------------------
// CSR build step 2: exclusive prefix scan over counts[N] -> offsets[N+1],
// plus a cursor copy for the scatter pass. Single block, 1024 threads,
// two-level scan (per-thread chunk sums + Hillis-Steele over 1024 partials).
// ---------------------------------------------------------------------------
__global__ void __launch_bounds__(1024)
fgcn_scan(const int* __restrict__ counts, int* __restrict__ offsets,
          int* __restrict__ cursor, int N, int E) {
    __shared__ int sdata[1024];
    const int tid = threadIdx.x;
    const int chunk = (N + 1023) / 1024;
    const int beg = tid * chunk;
    const int end = min(beg + chunk, N);

    int mysum = 0;
    for (int i = beg; i < end; ++i) mysum += counts[i];
    sdata[tid] = mysum;
    __syncthreads();

    for (int off = 1; off < 1024; off <<= 1) {
        int t = (tid >= off) ? sdata[tid - off] : 0;
        __syncthreads();
        sdata[tid] += t;
        __syncthreads();
    }
    int running = sdata[tid] - mysum;   // exclusive base for this chunk

    for (int i = beg; i < end; ++i) {
        offsets[i] = running;
        cursor[i]  = running;
        running += counts[i];
    }
    if (tid == 0) offsets[N] = E;
}

// ---------------------------------------------------------------------------
// CSR build step 3: scatter the edge's SOURCE NODE ID (not the edge id) into
// per-destination segments — the gather loop only ever needs src[e], so this
// removes one level of indirection from its critical load chain.
// ---------------------------------------------------------------------------
__global__ void __launch_bounds__(256)
fgcn_scatter_src(const int* __restrict__ src, const int* __restrict__ dst,
                 int* __restrict__ cursor, int* __restrict__ srcs, int E) {
    int e = blockIdx.x * 256 + threadIdx.x;
    if (e >= E) return;
    int pos = atomicAdd(&cursor[dst[e]], 1);
    srcs[pos] = src[e];
}

// ---------------------------------------------------------------------------
// Kernel 1: h2 = h @ W_node + b_node        (N x 128) = (N x 128)(128 x 128)
// Block: 256 threads = 8 waves; block = 16 rows, wave w = 16 cols.
// A-tile in LDS (padded stride 132: bank-conflict-free, 16B aligned);
// B from packed layout: one global_load_b64 per WMMA.
// ---------------------------------------------------------------------------
__global__ void __launch_bounds__(256)
fgcn_node_gemm(const float* __restrict__ h, const float2* __restrict__ Bp,
               const float* __restrict__ bias, float* __restrict__ h2) {
    __shared__ float ldsA[16 * 132];

    const int rowBase = blockIdx.x * 16;
    const int t = threadIdx.x;

    {
        const int r = t >> 4;
        const int c = (t & 15) * 8;
        const float4* s = (const float4*)(h + (size_t)(rowBase + r) * DFEAT + c);
        float4 v0 = s[0];
        float4 v1 = s[1];
        float4* d = (float4*)&ldsA[r * 132 + c];
        d[0] = v0;
        d[1] = v1;
    }
    __syncthreads();

    const int wave = t >> 5;
    const int lane = t & 31;
    const int half = lane >> 4;
    const int m    = lane & 15;
    const int col  = wave * 16 + m;
    const float2* bptr = Bp + (size_t)col * 2 + half;   // + g*256 per group

    v8f acc = {};
#pragma unroll
    for (int g = 0; g < DFEAT / 4; ++g) {
        const int k = 4 * g;
        v2f a, b;
        a[0] = ldsA[m * 132 + k + 2 * half];
        a[1] = ldsA[m * 132 + k + 2 * half + 1];
        float2 bv = bptr[(size_t)g * 2 * DFEAT];
        b[0] = bv.x;
        b[1] = bv.y;
        acc = __builtin_amdgcn_wmma_f32_16x16x4_f32(
            false, a, false, b, (short)0, acc, false, false);
    }

    const float bv = bias[col];
#pragma unroll
    for (int i = 0; i < 8; ++i) {
        const int row = rowBase + i + 8 * half;   // VGPR i: M=i (lo) / i+8 (hi)
        h2[(size_t)row * DFEAT + col] = acc[i] + bv;
    }
}

// ---------------------------------------------------------------------------
// Kernel 2: gather-reduce per destination node (atomic-free).
//   h_sum[n] = deg(n)*att[n]*h2[n] + sum_{j in segment(n)} att[s_j]*h2[s_j]
// (dst-side contribution factored algebraically: deg * att[n] * h2[n].)
// One wave per node; lane l owns floats 4l..4l+3. Per edge: one sequential
// 4B read (srcs), one cached 4B read (att), one 512B/wave row gather (h2).
// ---------------------------------------------------------------------------
__global__ void __launch_bounds__(256)
fgcn_gather_reduce(const float* __restrict__ h2, const float* __restrict__ att,
                   const int* __restrict__ srcs, const int* __restrict__ offsets,
                   float* __restrict__ h_sum, int N) {
    const int n = blockIdx.x * 8 + (threadIdx.x >> 5);
    const int lane = threadIdx.x & 31;
    if (n >= N) return;

    const int beg = offsets[n];
    const int end = offsets[n + 1];

    const float an = att[n];
    const float4 hn = ((const float4*)(h2 + (size_t)n * DFEAT))[lane];
    const float w0 = (float)(end - beg) * an;

    float4 acc;
    acc.x = w0 * hn.x;
    acc.y = w0 * hn.y;
    acc.z = w0 * hn.z;
    acc.w = w0 * hn.w;

    for (int j = beg; j < end; ++j) {
        const int s = srcs[j];
        const float as = att[s];
        const float4 hs = ((const float4*)(h2 + (size_t)s * DFEAT))[lane];
        acc.x += as * hs.x;
        acc.y += as * hs.y;
        acc.z += as * hs.z;
        acc.w += as * hs.w;
    }
    ((float4*)(h_sum + (size_t)n * DFEAT))[lane] = acc;
}

// ---------------------------------------------------------------------------
// Kernel 3: out = relu(concat(h2, h_sum) @ W_apply + b_apply), K = 256.
// Same tiling; LDS A-tile 16x256 (padded stride 260); packed B.
// ---------------------------------------------------------------------------
__global__ void __launch_bounds__(256)
fgcn_apply_gemm(const float* __restrict__ h2, const float* __restrict__ h_sum,
                const float2* __restrict__ Bp, const float* __restrict__ bias,
                float* __restrict__ out) {
    __shared__ float ldsA[16 * 260];

    const int rowBase = blockIdx.x * 16;
    const int t = threadIdx.x;

    {
        const int r  = t >> 4;
        const int cg = (t & 15) * 16;
        const float* srow = (cg < DFEAT)
            ? (h2    + (size_t)(rowBase + r) * DFEAT + cg)
            : (h_sum + (size_t)(rowBase + r) * DFEAT + (cg - DFEAT));
        const float4* s = (const float4*)srow;
        float4 v0 = s[0], v1 = s[1], v2 = s[2], v3 = s[3];
        float4* d = (float4*)&ldsA[r * 260 + cg];
        d[0] = v0; d[1] = v1; d[2] = v2; d[3] = v3;
    }
    __syncthreads();

    const int wave = t >> 5;
    const int lane = t & 31;
    const int half = lane >> 4;
    const int m    = lane & 15;
    const int col  = wave * 16 + m;
    const float2* bptr = Bp + (size_t)col * 2 + half;

    v8f acc = {};
#pragma unroll
    for (int g = 0; g < (2 * DFEAT) / 4; ++g) {
        const int k = 4 * g;
        v2f a, b;
        a[0] = ldsA[m * 260 + k + 2 * half];
        a[1] = ldsA[m * 260 + k + 2 * half + 1];
        float2 bv = bptr[(size_t)g * 2 * DFEAT];
        b[0] = bv.x;
        b[1] = bv.y;
        acc = __builtin_amdgcn_wmma_f32_16x16x4_f32(
            false, a, false, b, (short)0, acc, false, false);
    }

    const float bvv = bias[col];
#pragma unroll
    for (int i = 0; i < 8; ++i) {
        const int row = rowBase + i + 8 * half;
        out[(size_t)row * DFEAT + col] = fmaxf(acc[i] + bvv, 0.f);
    }
}

// ---------------------------------------------------------------------------
// launch
// ---------------------------------------------------------------------------
extern "C" void kernel_launch(void* const* d_in, const int* in_sizes, int n_in,
                              void* d_out, int out_size, void* d_ws, size_t ws_size,
                              hipStream_t stream) {
    const float* h       = (const float*)d_in[0];  // [N,128]
    const float* att     = (const float*)d_in[1];  // [N,1]
    const int*   src     = (const int*)  d_in[2];  // [E]
    const int*   dst     = (const int*)  d_in[3];  // [E]
    const float* W_node  = (const float*)d_in[4];  // [128,128]
    const float* b_node  = (const float*)d_in[5];  // [128]
    const float* W_apply = (const float*)d_in[6];  // [256,128]
    const float* b_apply = (const float*)d_in[7];  // [128]
    float*       out     = (float*)d_out;          // [N,128]

    const int N = in_sizes[0] / DFEAT;
    const int E = in_sizes[2];

    // workspace layout
    float*  h2     = (float*)d_ws;                         // N*128 f32
    float*  h_sum  = h2 + (size_t)N * DFEAT;               // N*128 f32
    float2* Wn_p   = (float2*)(h_sum + (size_t)N * DFEAT); // 8192  float2
    float2* Wa_p   = Wn_p + (DFEAT / 4) * DFEAT * 2;       // 16384 float2
    int*    counts = (int*)(Wa_p + (2 * DFEAT / 4) * DFEAT * 2); // N
    int*    offs   = counts + N;                           // N+1 (+pad)
    int*    cursor = offs + N + 4;                         // N
    int*    srcs   = cursor + N;                           // E (sorted src ids)

    // one-shot weight repacks (tiny)
    fgcn_pack_w<<<((DFEAT / 4) * DFEAT * 2 + 255) / 256, 256, 0, stream>>>(
        W_node, Wn_p, DFEAT);
    fgcn_pack_w<<<((2 * DFEAT / 4) * DFEAT * 2 + 255) / 256, 256, 0, stream>>>(
        W_apply, Wa_p, 2 * DFEAT);

    // CSR build: zero counts -> histogram -> scan -> scatter src ids
    fgcn_zero_i32<<<(N + 255) / 256, 256, 0, stream>>>(counts, N);
    fgcn_histogram<<<(E + 255) / 256, 256, 0, stream>>>(dst, counts, E);
    fgcn_scan<<<1, 1024, 0, stream>>>(counts, offs, cursor, N, E);
    fgcn_scatter_src<<<(E + 255) / 256, 256, 0, stream>>>(src, dst, cursor, srcs, E);

    // h2 = h @ W_node + b_node
    fgcn_node_gemm<<<N / 16, 256, 0, stream>>>(h, Wn_p, b_node, h2);

    // atomic-free per-node gather reduction
    fgcn_gather_reduce<<<(N + 7) / 8, 256, 0, stream>>>(
        h2, att, srcs, offs, h_sum, N);

    // out = relu(concat(h2, h_sum) @ W_apply + b_apply)
    fgcn_apply_gemm<<<N / 16, 256, 0, stream>>>(h2, h_sum, Wa_p, b_apply, out);
}